// MaskedGINE_54958401520128
// MI455X (gfx1250) — compile-verified
//
#include <hip/hip_runtime.h>
#include <cstddef>
#include <cstdint>

// ---------------------------------------------------------------------------
// MaskedGINE forward for MI455X (gfx1250), fp32 via V_WMMA_F32_16X16X4_F32.
// GEMM FLOPs ~1.24e11 fp32; scatter traffic ~1.5GB (@23.3TB/s -> ~0.1ms).
// fp32 WMMA keeps reference numerics while using the matrix pipe.
// h/agg/tmp (~180MB) fit in the 192MB L2, so scatter atomics resolve in L2.
// W tiles are staged LDS via GLOBAL_LOAD_ASYNC_TO_LDS_B128 (ASYNCcnt),
// double-buffered so staging overlaps the WMMA k-loop.
// ---------------------------------------------------------------------------

typedef __attribute__((ext_vector_type(2))) float v2f;
typedef __attribute__((ext_vector_type(8))) float v8f;

#define NND    50000
#define NED    125000
#define DIM    300
#define NVOCAB 87
#define KANCH  6
#define NLAY   5
#define BN_EPS 1e-5f

#define BM   128   // rows per block  (8 waves x 16)
#define BNT  64    // cols per block  (4 wmma tiles per wave)
#define KC   100   // K chunk staged in LDS (300 = 3*100, no tail)

// AMODE: 0 = plain A[M x 300]; 1 = A generated on-the-fly: relu(attr.W1 + b1);
//        2 = A + A2 (elementwise, fuses h + agg)
// EPI:   0 = acc + bias; 1 = relu(acc + bias); 2 = relu(BN(acc + bias));
//        3 = acc + bias + atom_emb[z[row]]
template<int AMODE, int NATTR, int EPI>
__global__ __launch_bounds__(256)
void gemm_wmma(const float* __restrict__ A, const float* __restrict__ A2,
               const float* __restrict__ p0, const float* __restrict__ p1,
               const float* __restrict__ p2, int attrStride,
               const float* __restrict__ W1f, const float* __restrict__ b1f,
               const float* __restrict__ W, int ldb,
               const float* __restrict__ bias,
               const float* __restrict__ gamma, const float* __restrict__ beta,
               const float* __restrict__ mean, const float* __restrict__ var,
               const int* __restrict__ z, const float* __restrict__ emb,
               float* __restrict__ Cout, int ldc, int M, int N)
{
  __shared__ __align__(16) float sW[2][KC * BNT];

  const int tid  = threadIdx.x;
  const int lane = tid & 31;
  const int wave = tid >> 5;
  const int hi   = lane >> 4;   // 0: lanes 0-15, 1: lanes 16-31
  const int lo   = lane & 15;
  const int rowA    = blockIdx.x * BM + wave * 16 + lo;
  const int colBase = blockIdx.y * BNT;
  const int K = DIM;

  // Fast path: whole 64-col panel in range and ldb 16B-friendly -> async B128.
  const bool fastStage = ((ldb & 3) == 0) && (colBase + BNT <= N);

  // Stage W[k0..k0+KC) x [colBase..colBase+64) into sW[buf].
  auto stage = [&](int buf, int k0) {
    if (fastStage) {
      for (int i = tid; i < KC * (BNT / 4); i += 256) {
        const int r = i >> 4, c4 = i & 15;
        const float* g = W + (size_t)(k0 + r) * ldb + colBase + c4 * 4;
        const unsigned ldsAddr =
            (unsigned)(uintptr_t)(&sW[buf][r * BNT + c4 * 4]);
        // 16B global -> LDS, no VGPR round trip; tracked by ASYNCcnt.
        asm volatile("global_load_async_to_lds_b128 %0, %1, off"
                     :: "v"(ldsAddr), "v"(g) : "memory");
      }
    } else {
      for (int i = tid; i < KC * BNT; i += 256) {
        const int r = i >> 6, c = i & 63;
        const int col = colBase + c;
        sW[buf][i] = (col < N) ? W[(size_t)(k0 + r) * ldb + col] : 0.f;
      }
    }
  };

  // per-row scalars for the fused tiny front layer (AMODE 1)
  float s0 = 0.f, s1 = 0.f, s2 = 0.f;
  if (AMODE == 1) {
    if (rowA < M) {
      s0 = p0[(size_t)rowA * attrStride];
      s1 = p1[(size_t)rowA * attrStride];
      if (NATTR == 3) s2 = p2[(size_t)rowA * attrStride];
    }
  }

  v8f acc[4];
#pragma unroll
  for (int t = 0; t < 4; ++t)
#pragma unroll
    for (int j = 0; j < 8; ++j) acc[t][j] = 0.f;

  // prologue: stage chunk 0, wait, barrier
  stage(0, 0);
  if (fastStage) asm volatile("s_wait_asynccnt 0" ::: "memory");
  __syncthreads();

  const int NCHUNK = K / KC;  // 3
  for (int c = 0; c < NCHUNK; ++c) {
    const int cur = c & 1;
    const int k0  = c * KC;

    if (c + 1 < NCHUNK) {
      // overlap: stage next chunk into the other buffer while computing
      stage(cur ^ 1, k0 + KC);
      if (AMODE != 1 && rowA < M)  // pull next A chunk toward L0/L2
        __builtin_prefetch(A + (size_t)rowA * K + k0 + KC, 0, 0);
    }

    for (int kk = 0; kk < KC; kk += 4) {
      const int kA = k0 + kk + 2 * hi;  // even -> 8B-aligned float2 loads
      v2f a;
      if (AMODE == 0) {
        if (rowA < M) a = *(const v2f*)(A + (size_t)rowA * K + kA);
        else          { a.x = 0.f; a.y = 0.f; }
      } else if (AMODE == 2) {
        if (rowA < M) {
          v2f x = *(const v2f*)(A  + (size_t)rowA * K + kA);
          v2f y = *(const v2f*)(A2 + (size_t)rowA * K + kA);
          a = x + y;
        } else { a.x = 0.f; a.y = 0.f; }
      } else {  // AMODE 1: relu(attr . W1 + b1), K_front = NATTR (2 or 3)
        float ax = s0 * W1f[kA]     + s1 * W1f[K + kA]     + b1f[kA];
        float ay = s0 * W1f[kA + 1] + s1 * W1f[K + kA + 1] + b1f[kA + 1];
        if (NATTR == 3) {
          ax += s2 * W1f[2 * K + kA];
          ay += s2 * W1f[2 * K + kA + 1];
        }
        a.x = ax > 0.f ? ax : 0.f;
        a.y = ay > 0.f ? ay : 0.f;
      }

#pragma unroll
      for (int t = 0; t < 4; ++t) {
        v2f b;
        const int cc = t * 16 + lo;
        b.x = sW[cur][(kk + 2 * hi)     * BNT + cc];
        b.y = sW[cur][(kk + 2 * hi + 1) * BNT + cc];
        // D(f32 16x16) = A(16x4 f32) * B(4x16 f32) + C
        acc[t] = __builtin_amdgcn_wmma_f32_16x16x4_f32(
            false, a, false, b, (short)0, acc[t], false, false);
      }
    }

    if (c + 1 < NCHUNK) {
      // next buffer fully landed (own asyncs), then block-wide handshake
      if (fastStage) asm volatile("s_wait_asynccnt 0" ::: "memory");
      __syncthreads();
    }
  }

  // epilogue: C VGPR j holds rows (j, j+8) for lane halves; N = lane&15
  const int tileRow = blockIdx.x * BM + wave * 16;
#pragma unroll
  for (int t = 0; t < 4; ++t) {
    const int col = colBase + t * 16 + lo;
    if (col >= N) continue;
    const float bval = bias[col];
    float g = 0.f, bt = 0.f, mn = 0.f, inv = 0.f;
    if (EPI == 2) {
      g = gamma[col]; bt = beta[col]; mn = mean[col];
      inv = rsqrtf(var[col] + BN_EPS);
    }
#pragma unroll
    for (int j = 0; j < 8; ++j) {
      const int row = tileRow + j + 8 * hi;
      if (row >= M) continue;
      float v = acc[t][j] + bval;
      if (EPI == 1)      v = fmaxf(v, 0.f);
      else if (EPI == 2) { v = g * (v - mn) * inv + bt; v = fmaxf(v, 0.f); }
      else if (EPI == 3) v += emb[(size_t)z[row] * DIM + col];
      Cout[(size_t)row * ldc + col] = v;
    }
  }
}

// agg[dst] += relu(h[src] + edge_emb[e]); one edge per 128-thread block.
// unsafeAtomicAdd -> non-returning global_atomic_add_f32 (L2-resident).
__global__ __launch_bounds__(128)
void scatter_msgs(const float* __restrict__ h, const float* __restrict__ ee,
                  const int* __restrict__ eidx, float* __restrict__ agg)
{
  const int e = blockIdx.x;
  const int s = eidx[e];
  const int d = eidx[NED + e];
  const float* hs = h  + (size_t)s * DIM;
  const float* es = ee + (size_t)e * DIM;
  float*       ad = agg + (size_t)d * DIM;
  for (int k = threadIdx.x; k < DIM; k += 128) {
    float m = hs[k] + es[k];
    m = m > 0.f ? m : 0.f;
    unsafeAtomicAdd(&ad[k], m);
  }
}

extern "C" void kernel_launch(void* const* d_in, const int* in_sizes, int n_in,
                              void* d_out, int out_size, void* d_ws, size_t ws_size,
                              hipStream_t stream)
{
  (void)in_sizes; (void)n_in; (void)out_size; (void)ws_size;

  const int*   z        = (const int*)  d_in[0];
  const float* chir     = (const float*)d_in[1];
  const float* fchg     = (const float*)d_in[2];
  const int*   eidx     = (const int*)  d_in[3];
  const float* eattr    = (const float*)d_in[4];
  const float* atom_emb = (const float*)d_in[5];
  const float* nap_W1   = (const float*)d_in[6];
  const float* nap_b1   = (const float*)d_in[7];
  const float* nap_W2   = (const float*)d_in[8];
  const float* nap_b2   = (const float*)d_in[9];
  const float* ee_W1    = (const float*)d_in[10];
  const float* ee_b1    = (const float*)d_in[11];
  const float* ee_W2    = (const float*)d_in[12];
  const float* ee_b2    = (const float*)d_in[13];
  const float* gin_W1   = (const float*)d_in[14];
  const float* gin_b1   = (const float*)d_in[15];
  const float* gin_W2   = (const float*)d_in[16];
  const float* gin_b2   = (const float*)d_in[17];
  const float* bn_g     = (const float*)d_in[18];
  const float* bn_b     = (const float*)d_in[19];
  const float* bn_m     = (const float*)d_in[20];
  const float* bn_v     = (const float*)d_in[21];
  const float* head_Wz  = (const float*)d_in[22];
  const float* head_bz  = (const float*)d_in[23];
  const float* head_Wd  = (const float*)d_in[24];
  const float* head_bd  = (const float*)d_in[25];

  // workspace layout (floats): edge_emb | h | agg | tmp  (~330 MB total)
  float* ws       = (float*)d_ws;
  float* edge_emb = ws;
  float* h        = edge_emb + (size_t)NED * DIM;
  float* agg      = h   + (size_t)NND * DIM;
  float* tmp      = agg + (size_t)NND * DIM;

  const dim3 blk(256);
  auto grd = [](int M, int N) { return dim3((M + BM - 1) / BM, (N + BNT - 1) / BNT); };

  // h = atom_emb[z] + mlp2([chirality, formal_charge])
  gemm_wmma<1, 2, 3><<<grd(NND, DIM), blk, 0, stream>>>(
      nullptr, nullptr, chir, fchg, nullptr, 1,
      nap_W1, nap_b1, nap_W2, DIM, nap_b2,
      nullptr, nullptr, nullptr, nullptr, z, atom_emb,
      h, DIM, NND, DIM);

  // edge_emb = mlp2(edge_attr)
  gemm_wmma<1, 3, 0><<<grd(NED, DIM), blk, 0, stream>>>(
      nullptr, nullptr, eattr, eattr + 1, eattr + 2, 3,
      ee_W1, ee_b1, ee_W2, DIM, ee_b2,
      nullptr, nullptr, nullptr, nullptr, nullptr, nullptr,
      edge_emb, DIM, NED, DIM);

  for (int l = 0; l < NLAY; ++l) {
    hipMemsetAsync(agg, 0, sizeof(float) * (size_t)NND * DIM, stream);
    scatter_msgs<<<NED, 128, 0, stream>>>(h, edge_emb, eidx, agg);

    // tmp = relu((h + agg) @ W1 + b1)
    gemm_wmma<2, 0, 1><<<grd(NND, DIM), blk, 0, stream>>>(
        h, agg, nullptr, nullptr, nullptr, 0,
        nullptr, nullptr, gin_W1 + (size_t)l * DIM * DIM, DIM, gin_b1 + l * DIM,
        nullptr, nullptr, nullptr, nullptr, nullptr, nullptr,
        tmp, DIM, NND, DIM);

    // h = relu(BN(tmp @ W2 + b2))
    gemm_wmma<0, 0, 2><<<grd(NND, DIM), blk, 0, stream>>>(
        tmp, nullptr, nullptr, nullptr, nullptr, 0,
        nullptr, nullptr, gin_W2 + (size_t)l * DIM * DIM, DIM, gin_b2 + l * DIM,
        bn_g + l * DIM, bn_b + l * DIM, bn_m + l * DIM, bn_v + l * DIM,
        nullptr, nullptr, h, DIM, NND, DIM);
  }

  float* out = (float*)d_out;
  // logits = h @ head_Wz + bz  ([N, 87])
  gemm_wmma<0, 0, 0><<<grd(NND, NVOCAB), blk, 0, stream>>>(
      h, nullptr, nullptr, nullptr, nullptr, 0,
      nullptr, nullptr, head_Wz, NVOCAB, head_bz,
      nullptr, nullptr, nullptr, nullptr, nullptr, nullptr,
      out, NVOCAB, NND, NVOCAB);
  // dists = h @ head_Wd + bd  ([N, 6])
  gemm_wmma<0, 0, 0><<<grd(NND, KANCH), blk, 0, stream>>>(
      h, nullptr, nullptr, nullptr, nullptr, 0,
      nullptr, nullptr, head_Wd, KANCH, head_bd,
      nullptr, nullptr, nullptr, nullptr, nullptr, nullptr,
      out + (size_t)NND * KANCH * 0 + (size_t)NND * NVOCAB, KANCH, NND, KANCH);
}